// DifferentiableRasterizer_67242007986932
// MI455X (gfx1250) — compile-verified
//
#include <hip/hip_runtime.h>

typedef float v2f __attribute__((ext_vector_type(2)));
typedef float v8f __attribute__((ext_vector_type(8)));

#define EPSV      1e-15f
#define MAXN      0.99999f          // 1 - 1e-5
#define IMGSZ     256
#define NPIX      (IMGSZ * IMGSZ)   // 65536
#define NTAXA     1024
#define NCH       16
#define NB        4
#define MTOT      (NB * NCH)        // 64
#define CHUNK_K   64
#define PIX_PER_BLK 64
#define KSTR      66                // padded LDS stride (floats) per pixel column

// workspace layout (float offsets)
// Wtp: K-pair interleaved weights: Wtp[(t>>1)*128 + m*2 + (t&1)] = A[b,t]*F[t,c]
#define WT_OFF    0                       // Wtp[NTAXA/2][MTOT][2]
#define PX_OFF    (NTAXA * MTOT)          // projx[NTAXA]
#define PY_OFF    (PX_OFF + NTAXA)        // projy[NTAXA]
#define NI_OFF    (PY_OFF + NTAXA)        // negInv[NTAXA] = -1/(2*bw^2+eps)
#define IMG_OFF   (NI_OFF + NTAXA)        // image[MTOT][NPIX]

// ---------------------------------------------------------------------------
// Kernel 1: per-taxon preprocessing + weight matrix (single 1024-thread block)
// ---------------------------------------------------------------------------
__global__ __launch_bounds__(1024)
void setup_kernel(const float* __restrict__ A,       // [4][1024] abundances
                  const float* __restrict__ emb,     // [1024][16] embeddings
                  const float* __restrict__ F,       // [1024][16] annotations
                  const float* __restrict__ depths,  // [1024]
                  float* __restrict__ ws)
{
    __shared__ float lo[1024];
    __shared__ float hi[1024];
    const int t = threadIdx.x;

    // stereographic projection of first two embedding coords
    float x = emb[t * NCH + 0];
    float y = emb[t * NCH + 1];
    float r2 = fminf(x * x + y * y, MAXN * MAXN);
    float denom = fmaxf(1.0f - r2, EPSV);     // CURVATURE == 1
    float scale = 2.0f / denom;
    float px = x * scale;
    float py = y * scale;

    // global min/max over all 2048 projected coordinates
    lo[t] = fminf(px, py);
    hi[t] = fmaxf(px, py);
    __syncthreads();
    for (int s = 512; s > 0; s >>= 1) {
        if (t < s) {
            lo[t] = fminf(lo[t], lo[t + s]);
            hi[t] = fmaxf(hi[t], hi[t + s]);
        }
        __syncthreads();
    }
    float pmin = lo[0];
    float pmax = hi[0];
    __syncthreads();

    float prange = pmax - pmin;
    float sx = 0.0f, sy = 0.0f;
    if (prange > EPSV) {
        sx = 1.8f * (px - pmin) / prange - 0.9f;
        sy = 1.8f * (py - pmin) / prange - 0.9f;
    }

    // adaptive bandwidth from depths
    float d = depths[t];
    lo[t] = d;
    hi[t] = d;
    __syncthreads();
    for (int s = 512; s > 0; s >>= 1) {
        if (t < s) {
            lo[t] = fminf(lo[t], lo[t + s]);
            hi[t] = fmaxf(hi[t], hi[t + s]);
        }
        __syncthreads();
    }
    float dmin = lo[0];
    float dmax = hi[0];
    float nd = (dmax > dmin) ? (d - dmin) / (dmax - dmin) : 0.5f;
    float bw = 0.1f * (1.5f - nd);
    float negInv = -1.0f / (2.0f * bw * bw + EPSV);

    ws[PX_OFF + t] = sx;
    ws[PY_OFF + t] = sy;
    ws[NI_OFF + t] = negInv;

    // K-pair interleaved weights: Wtp[(t>>1)*128 + m*2 + (t&1)] = A[b,t]*F[t,c]
    const int base = WT_OFF + (t >> 1) * (MTOT * 2) + (t & 1);
#pragma unroll
    for (int b = 0; b < NB; ++b) {
        float ab = A[b * NTAXA + t];
#pragma unroll
        for (int c = 0; c < NCH; ++c) {
            ws[base + (b * NCH + c) * 2] = ab * F[t * NCH + c];
        }
    }
}

// ---------------------------------------------------------------------------
// Kernel 2: fused exp-kernel generation + 64x1024x65536 GEMM via f32 WMMA.
// 1024 blocks x 256 threads (8 waves). Each block owns 64 pixels.
// Wave w: Mtile = w>>1 (0..3), N-tile pair = (w&1)*2 + {0,1}.
// A-frag: one coalesced global_load_b64 (K-pair interleaved layout).
// B-frags: one ds_load_b64 each (K-major LDS tile, padded stride).
// ---------------------------------------------------------------------------
__global__ __launch_bounds__(256)
void raster_kernel(const float* __restrict__ Wtp,     // [512][64][2]
                   const float* __restrict__ projx,   // [1024]
                   const float* __restrict__ projy,   // [1024]
                   const float* __restrict__ negInv,  // [1024]
                   float* __restrict__ img)           // [64][65536]
{
    __shared__ float kb[PIX_PER_BLK * KSTR];   // K-major exp tile: kb[col*66 + k]

    const int tid    = threadIdx.x;
    const int lane   = tid & 31;      // wave32
    const int waveId = tid >> 5;      // 0..7
    const int Mtile  = waveId >> 1;   // 0..3
    const int Nt0    = (waveId & 1) * 2;  // 0 or 2
    const int half   = lane >> 4;     // 0: lanes 0-15, 1: lanes 16-31
    const int lane16 = lane & 15;

    const int pbase = blockIdx.x * PIX_PER_BLK;
    // pixel coordinates for the fill phase: cols lane and lane+32 of this tile
    const float gy  = -1.0f + (float)(pbase >> 8) * (2.0f / 255.0f);
    const float gx0 = -1.0f + (float)((pbase & (IMGSZ - 1)) + lane) * (2.0f / 255.0f);
    const float gx1 = gx0 + 32.0f * (2.0f / 255.0f);

    // per-wave operand base addresses
    const float* aBase = Wtp + (Mtile * 16 + lane16) * 2;           // + arow*64
    const float* b0Base = &kb[(Nt0 * 16 + lane16) * KSTR + half * 2];
    const float* b1Base = &kb[((Nt0 + 1) * 16 + lane16) * KSTR + half * 2];

    v8f acc0 = {0.f, 0.f, 0.f, 0.f, 0.f, 0.f, 0.f, 0.f};
    v8f acc1 = {0.f, 0.f, 0.f, 0.f, 0.f, 0.f, 0.f, 0.f};

    for (int kbase = 0; kbase < NTAXA; kbase += CHUNK_K) {
        // prefetch next chunk of Wtp into cache (global_prefetch_b8 path)
        if (kbase + CHUNK_K < NTAXA) {
            __builtin_prefetch(Wtp + (kbase + CHUNK_K) * MTOT + tid * 64, 0, 3);
        }
        __syncthreads();   // previous chunk's WMMA reads done before refill
        // fill 64(K) x 64(pix) exp tile, K-major per column:
        // thread (waveId, lane) covers rows waveId+8j, cols lane and lane+32
#pragma unroll
        for (int j = 0; j < 8; ++j) {
            int row = waveId + j * 8;
            int t = kbase + row;
            float qx = projx[t];
            float qy = projy[t];
            float ni = negInv[t];
            float dy  = gy - qy;
            float dy2 = dy * dy;
            float dx0 = gx0 - qx;
            float dx1 = gx1 - qx;
            kb[lane * KSTR + row]        = __expf((dx0 * dx0 + dy2) * ni);
            kb[(lane + 32) * KSTR + row] = __expf((dx1 * dx1 + dy2) * ni);
        }
        __syncthreads();

        // 16 WMMA K-steps of 4 over this chunk; one A-frag feeds two tiles
#pragma unroll
        for (int kk = 0; kk < CHUNK_K; kk += 4) {
            // A-frag (16x4 f32): lanes 0-15 hold K=kk,kk+1; lanes 16-31 hold K=kk+2,kk+3
            // arow is even -> {W[arow][m], W[arow+1][m]} contiguous in Wtp
            int arow = kbase + kk + half * 2;
            v2f a  = *(const v2f*)(aBase + arow * MTOT);
            // B-frags (4x16 f32): {kb[col][brow], kb[col][brow+1]} contiguous in LDS
            v2f b0 = *(const v2f*)(b0Base + kk);
            v2f b1 = *(const v2f*)(b1Base + kk);
            acc0 = __builtin_amdgcn_wmma_f32_16x16x4_f32(
                false, a, false, b0, (short)0, acc0, false, false);
            acc1 = __builtin_amdgcn_wmma_f32_16x16x4_f32(
                false, a, false, b1, (short)0, acc1, false, false);
        }
    }

    // C/D layout: VGPR r -> row (r + half*8), lane16 -> column
    const int m0 = Mtile * 16 + half * 8;
    const int pix0 = pbase + Nt0 * 16 + lane16;
    const int pix1 = pbase + (Nt0 + 1) * 16 + lane16;
#pragma unroll
    for (int r = 0; r < 8; ++r) {
        img[(size_t)(m0 + r) * NPIX + pix0] = acc0[r];
        img[(size_t)(m0 + r) * NPIX + pix1] = acc1[r];
    }
}

// ---------------------------------------------------------------------------
// Kernel 3: per-channel min/max normalization. 64 blocks x 256 threads.
// ---------------------------------------------------------------------------
__global__ __launch_bounds__(256)
void normalize_kernel(const float* __restrict__ img, float* __restrict__ out)
{
    __shared__ float slo[256];
    __shared__ float shi[256];
    const int ch = blockIdx.x;
    const float* p = img + (size_t)ch * NPIX;

    float lo = 3.402823466e38f;
    float hi = -3.402823466e38f;
    for (int i = threadIdx.x; i < NPIX; i += 256) {
        float v = p[i];
        lo = fminf(lo, v);
        hi = fmaxf(hi, v);
    }
    slo[threadIdx.x] = lo;
    shi[threadIdx.x] = hi;
    __syncthreads();
    for (int s = 128; s > 0; s >>= 1) {
        if (threadIdx.x < s) {
            slo[threadIdx.x] = fminf(slo[threadIdx.x], slo[threadIdx.x + s]);
            shi[threadIdx.x] = fmaxf(shi[threadIdx.x], shi[threadIdx.x + s]);
        }
        __syncthreads();
    }
    float cmin = slo[0];
    float crange = shi[0] - cmin;
    bool ok = crange > EPSV;
    float inv = 1.0f / (crange + EPSV);
    for (int i = threadIdx.x; i < NPIX; i += 256) {
        float v = p[i];
        out[(size_t)ch * NPIX + i] = ok ? (v - cmin) * inv : 0.0f;
    }
}

// ---------------------------------------------------------------------------
extern "C" void kernel_launch(void* const* d_in, const int* in_sizes, int n_in,
                              void* d_out, int out_size, void* d_ws, size_t ws_size,
                              hipStream_t stream)
{
    const float* abundances = (const float*)d_in[0];  // [4][1024]
    const float* embeddings = (const float*)d_in[1];  // [1024][16]
    const float* annot      = (const float*)d_in[2];  // [1024][16]
    const float* depths     = (const float*)d_in[3];  // [1024]
    float* out = (float*)d_out;                       // [4][16][256][256]
    float* ws  = (float*)d_ws;

    setup_kernel<<<1, 1024, 0, stream>>>(abundances, embeddings, annot, depths, ws);

    raster_kernel<<<NPIX / PIX_PER_BLK, 256, 0, stream>>>(
        ws + WT_OFF, ws + PX_OFF, ws + PY_OFF, ws + NI_OFF, ws + IMG_OFF);

    normalize_kernel<<<MTOT, 256, 0, stream>>>(ws + IMG_OFF, out);
}